// TacotronDuration_75677323755897
// MI455X (gfx1250) — compile-verified
//
#include <hip/hip_runtime.h>
#include <hip/hip_bf16.h>
#include <math.h>

// ---------------------------------------------------------------------------
// Problem constants (reference): B=16, K=512, H=F=256, TMAX=512, DW=16, DC=4
// ---------------------------------------------------------------------------
#define BB   16
#define KK   512
#define FF   256
#define TT   512
#define DWV  16
#define DCV  4

typedef __attribute__((ext_vector_type(2))) float v2f;
typedef __attribute__((ext_vector_type(8))) float v8f;

// CDNA5 fp32 WMMA: D(16x16,f32) = A(16x4,f32) x B(4x16,f32) + C
#define WMMA_F32(a, b, c) \
    __builtin_amdgcn_wmma_f32_16x16x4_f32(false, (a), false, (b), (short)0, (c), false, false)

__device__ __forceinline__ float swishf(float x) { return x / (1.f + expf(-x)); }

__device__ __forceinline__ float wred16(float v) {  // reduce within 16-lane half
    v += __shfl_xor(v, 1, 32); v += __shfl_xor(v, 2, 32);
    v += __shfl_xor(v, 4, 32); v += __shfl_xor(v, 8, 32);
    return v;
}
__device__ __forceinline__ float wred32(float v) {  // full wave32 reduce
    v += __shfl_xor(v, 16, 32);
    return wred16(v);
}
__device__ __forceinline__ float wmax32(float v) {
    for (int o = 16; o; o >>= 1) v = fmaxf(v, __shfl_xor(v, o, 32));
    return v;
}

// ---------------------------------------------------------------------------
// K0: one-shot weight transpose  w(r=tap*Cin+cin, f) -> wT(f, r)
// so a B-fragment lane reads two cin-adjacent values as ONE aligned b64.
// ---------------------------------------------------------------------------
__global__ __launch_bounds__(256) void transpose_w_kernel(
    const float* __restrict__ in, float* __restrict__ out, int R, int C)
{
    const int idx = blockIdx.x * 256 + threadIdx.x;     // idx = f*R + r
    if (idx < R * C) {
        const int f = idx / R, r = idx - f * R;
        out[idx] = in[r * C + f];
    }
}

// ---------------------------------------------------------------------------
// K1/K2: fused  conv1d(KS=3, 256->256) + ReLU + LayerNorm  [+ softplus head]
// grid = B*K/16 blocks of 512 threads (16 waves); wave w owns output N-tile w.
// A: im2col'd from an 18x256 LDS stage (row stride 258 -> bank-conflict-free,
// b64-aligned).  B: transposed weights, one b64 per lane per K-chunk.
// ---------------------------------------------------------------------------
#define XS_STRIDE 258
template <bool COMPUTE_D>
__global__ __launch_bounds__(512) void conv_relu_ln_kernel(
    const float* __restrict__ xin,   // (B,K,256)
    const float* __restrict__ wT,    // (256, 768) transposed weights
    const float* __restrict__ bias,  // (256)
    const float* __restrict__ ln_g, const float* __restrict__ ln_b,
    float* __restrict__ out,         // (B,K,256)
    const float* __restrict__ lin_w, const float* __restrict__ lin_b,
    const int* __restrict__ src_mask, float* __restrict__ d_out)
{
    __shared__ float xs[18 * XS_STRIDE];
    __shared__ float hs[16 * 260];
    const int tid = threadIdx.x;
    const int blk = blockIdx.x;
    const int b  = blk >> 5;                // 32 position-tiles per batch
    const int k0 = (blk & 31) << 4;

    for (int idx = tid; idx < 18 * 256; idx += 512) {
        const int row = idx >> 8, c = idx & 255;
        const int kk = k0 - 1 + row;        // 'SAME' zero padding per batch
        xs[row * XS_STRIDE + c] =
            (kk >= 0 && kk < KK) ? xin[((size_t)b * KK + kk) * 256 + c] : 0.f;
    }
    __syncthreads();

    const int wv = tid >> 5, lane = tid & 31;
    const int half = lane >> 4, ln16 = lane & 15;
    const int n0 = wv << 4;
    const float* wrow = wT + (size_t)(n0 + ln16) * 768;   // contiguous stream/lane

    v8f acc = {};
    #pragma unroll 1
    for (int tap = 0; tap < 3; ++tap) {
        const float* xrw = xs + (ln16 + tap) * XS_STRIDE;
        const float* wtp = wrow + tap * 256;
        for (int c0 = 0; c0 < 256; c0 += 4) {
            const int ca = c0 + (half << 1);
            const v2f a  = *(const v2f*)(xrw + ca);       // ds_load_b64
            const v2f bb = *(const v2f*)(wtp + ca);       // global_load_b64
            acc = WMMA_F32(a, bb, acc);
        }
    }

    const float bi = bias[n0 + ln16];
    #pragma unroll
    for (int r = 0; r < 8; ++r) {                   // D: vgpr r -> row r / r+8
        const int row = r + (half << 3);
        hs[row * 260 + n0 + ln16] = fmaxf(acc[r] + bi, 0.f);
    }
    __syncthreads();

    // LayerNorm over F=256: wave wv normalizes row wv (8 elems / lane).
    const int row = wv;
    float vals[8], sum = 0.f, sq = 0.f;
    #pragma unroll
    for (int i = 0; i < 8; ++i) {
        const float v = hs[row * 260 + lane * 8 + i];
        vals[i] = v; sum += v; sq += v * v;
    }
    sum = wred32(sum); sq = wred32(sq);
    const float mean = sum * (1.f / 256.f);
    const float var  = sq * (1.f / 256.f) - mean * mean;
    const float inv  = rsqrtf(var + 1e-5f);

    float dpart = 0.f;
    #pragma unroll
    for (int i = 0; i < 8; ++i) {
        const int f = lane * 8 + i;
        const float v = (vals[i] - mean) * inv * ln_g[f] + ln_b[f];
        out[((size_t)b * KK + k0 + row) * 256 + f] = v;
        if (COMPUTE_D) dpart += v * lin_w[f];
    }
    if (COMPUTE_D) {
        dpart = wred32(dpart);
        if (lane == 0) {
            const float z  = dpart + lin_b[0];
            const float sp = fmaxf(z, 0.f) + log1pf(expf(-fabsf(z)));  // softplus
            d_out[b * KK + k0 + row] = src_mask[b * KK + k0 + row] ? 0.f : sp;
        }
    }
}

// ---------------------------------------------------------------------------
// K3: per-batch exclusive scan of d, Tlen = rint(sum d) (RNE == jnp.round),
// mel_mask written as 0/1 floats.  grid = B, block = 512.
// ---------------------------------------------------------------------------
__global__ __launch_bounds__(512) void scan_kernel(
    const float* __restrict__ d, float* __restrict__ s_out, float* __restrict__ mel_out)
{
    __shared__ float buf[512];
    const int b = blockIdx.x, t = threadIdx.x;
    const float dv = d[b * KK + t];
    buf[t] = dv;
    __syncthreads();
    for (int off = 1; off < 512; off <<= 1) {
        float tmp = (t >= off) ? buf[t - off] : 0.f;
        __syncthreads();
        buf[t] += tmp;
        __syncthreads();
    }
    s_out[b * KK + t] = buf[t] - dv;                // exclusive cumsum
    const float Tlen = rintf(buf[511]);
    mel_out[b * TT + t] = ((float)t >= Tlen) ? 1.f : 0.f;
}

// ---------------------------------------------------------------------------
// K4: up-sample branches.  V1 = swish(LN16(convw(V))) via WMMA, transposed
// weights, mask-multiplied (non-divergent) boundary loads.
// V2 = swish(LN4(convc(V))) via VALU with float4 weight loads.
// ---------------------------------------------------------------------------
__global__ __launch_bounds__(256) void upbranch_kernel(
    const float* __restrict__ V,
    const float* __restrict__ wTw, const float* __restrict__ wbv,  // (16,768), (16)
    const float* __restrict__ lnw_g, const float* __restrict__ lnw_b,
    const float* __restrict__ wc, const float* __restrict__ cbv,   // (3,256,4), (4)
    const float* __restrict__ lnc_g, const float* __restrict__ lnc_b,
    float* __restrict__ V1, float* __restrict__ V2)
{
    const int wv = threadIdx.x >> 5, lane = threadIdx.x & 31;
    const int tile = blockIdx.x * 8 + wv;
    const int b = tile >> 5, k0 = (tile & 31) << 4;
    const int half = lane >> 4, ln16 = lane & 15;
    const float* Vb = V + (size_t)b * KK * 256;
    const float* wrow = wTw + (size_t)ln16 * 768;

    // ---- convw (256 -> 16) via WMMA; unconditional clamped loads * mask ----
    v8f acc = {};
    #pragma unroll 1
    for (int tap = 0; tap < 3; ++tap) {
        const int kr = k0 + ln16 - 1 + tap;
        const int krc = kr < 0 ? 0 : (kr > KK - 1 ? KK - 1 : kr);
        const float msk = (kr >= 0 && kr < KK) ? 1.f : 0.f;
        const float* xr = Vb + (size_t)krc * 256;
        const float* wtp = wrow + tap * 256;
        for (int c0 = 0; c0 < 256; c0 += 4) {
            const int ca = c0 + (half << 1);
            v2f a = *(const v2f*)(xr + ca);
            a.x *= msk; a.y *= msk;
            const v2f bb = *(const v2f*)(wtp + ca);
            acc = WMMA_F32(a, bb, acc);
        }
    }
    const float g16 = lnw_g[ln16], be16 = lnw_b[ln16], bo16 = wbv[ln16];
    #pragma unroll
    for (int r = 0; r < 8; ++r) {
        float v = acc[r] + bo16;                    // 16 dims of one row, across a half
        const float mean = wred16(v) * (1.f / 16.f);
        const float var  = wred16(v * v) * (1.f / 16.f) - mean * mean;
        float nv = (v - mean) * rsqrtf(var + 1e-5f) * g16 + be16;
        nv = swishf(nv);
        const int row = r + (half << 3);
        V1[((size_t)b * KK + k0 + row) * DWV + ln16] = nv;
    }

    // ---- convc (256 -> 4): halves split cin then combine; no divergence ----
    float a0 = 0.f, a1 = 0.f, a2 = 0.f, a3 = 0.f;
    #pragma unroll 1
    for (int tap = 0; tap < 3; ++tap) {
        const int kr = k0 + ln16 - 1 + tap;
        const int krc = kr < 0 ? 0 : (kr > KK - 1 ? KK - 1 : kr);
        const float msk = (kr >= 0 && kr < KK) ? 1.f : 0.f;
        const float* xr = Vb + (size_t)krc * 256;
        const float* wt = wc + (size_t)tap * 256 * 4;
        for (int c = half * 128; c < half * 128 + 128; ++c) {
            const float xv = xr[c] * msk;
            const float4 w4 = *(const float4*)(wt + c * 4);   // 16B aligned
            a0 += xv * w4.x; a1 += xv * w4.y; a2 += xv * w4.z; a3 += xv * w4.w;
        }
    }
    a0 += __shfl_xor(a0, 16, 32); a1 += __shfl_xor(a1, 16, 32);
    a2 += __shfl_xor(a2, 16, 32); a3 += __shfl_xor(a3, 16, 32);
    if (half == 0) {
        a0 += cbv[0]; a1 += cbv[1]; a2 += cbv[2]; a3 += cbv[3];
        const float mean = (a0 + a1 + a2 + a3) * 0.25f;
        const float var  = ((a0 - mean) * (a0 - mean) + (a1 - mean) * (a1 - mean) +
                            (a2 - mean) * (a2 - mean) + (a3 - mean) * (a3 - mean)) * 0.25f;
        const float inv = rsqrtf(var + 1e-5f);
        float* o = V2 + ((size_t)b * KK + k0 + ln16) * DCV;
        o[0] = swishf((a0 - mean) * inv * lnc_g[0] + lnc_b[0]);
        o[1] = swishf((a1 - mean) * inv * lnc_g[1] + lnc_b[1]);
        o[2] = swishf((a2 - mean) * inv * lnc_g[2] + lnc_b[2]);
        o[3] = swishf((a3 - mean) * inv * lnc_g[3] + lnc_b[3]);
    }
}

// ---------------------------------------------------------------------------
// K5: Wraw[b,k,t] = mlp1_3( swish(mlp1_2( swish(mlp1_1( S+E+V1 )) )) )
// One wave per (b,k); 32 t-tiles of 16; each layer = 4 chained
// v_wmma_f32_16x16x4_f32.  D->A turnaround via per-wave LDS tile (stride 18:
// even for b64 alignment, conflict-free mod 64), s_wait_dscnt guarded.
// ---------------------------------------------------------------------------
#define WL_STRIDE 18
__global__ __launch_bounds__(256) void wgrid_kernel(
    const float* __restrict__ V1, const float* __restrict__ dvec,
    const float* __restrict__ svec, const float* __restrict__ mel,
    const int* __restrict__ src_mask,
    const float* __restrict__ w1, const float* __restrict__ b1,
    const float* __restrict__ w2, const float* __restrict__ b2,
    const float* __restrict__ w3, const float* __restrict__ b3,
    float* __restrict__ Wraw)
{
    __shared__ float lds[8][16 * WL_STRIDE];
    const int wv = threadIdx.x >> 5, lane = threadIdx.x & 31;
    const int half = lane >> 4, ln16 = lane & 15;
    const int row = blockIdx.x * 8 + wv;            // b*K + k
    const int b = row >> 9;
    float* L = &lds[wv][0];

    v2f w1f[4], w2f[4];
    #pragma unroll
    for (int cc = 0; cc < 4; ++cc) {
        const int ca = cc * 4 + (half << 1);
        w1f[cc].x = w1[ca * 16 + ln16]; w1f[cc].y = w1[(ca + 1) * 16 + ln16];
        w2f[cc].x = w2[ca * 16 + ln16]; w2f[cc].y = w2[(ca + 1) * 16 + ln16];
    }
    const float bias1 = b1[ln16], bias2 = b2[ln16];
    float w3r[16], v1c[16];
    #pragma unroll
    for (int i = 0; i < 16; ++i) { w3r[i] = w3[i]; v1c[i] = V1[(size_t)row * DWV + i]; }
    const float b3v = b3[0];
    const float sk = svec[row], ek = sk + dvec[row];
    const int   srcm = src_mask[row];
    const float* melb = mel + (size_t)b * TT;
    float* Wout = Wraw + (size_t)row * TT;

    #pragma unroll 1
    for (int t0 = 0; t0 < TT; t0 += 16) {
        const int t = t0 + ln16;                    // A-frag row m = ln16
        const float pos = (float)(t + 1);
        const bool inv = srcm || (melb[t] != 0.f);
        const float se = inv ? 0.f : ((pos - sk) + (ek - pos));  // ref order

        v8f acc = {};
        #pragma unroll
        for (int cc = 0; cc < 4; ++cc) {
            const int ca = cc * 4 + (half << 1);
            v2f a; a.x = se + v1c[ca]; a.y = se + v1c[ca + 1];
            acc = WMMA_F32(a, w1f[cc], acc);
        }
        #pragma unroll
        for (int r = 0; r < 8; ++r)
            L[(r + (half << 3)) * WL_STRIDE + ln16] = swishf(acc[r] + bias1);
        asm volatile("s_wait_dscnt 0" ::: "memory");

        v8f acc2 = {};
        #pragma unroll
        for (int cc = 0; cc < 4; ++cc) {
            const int ca = cc * 4 + (half << 1);
            const v2f a = *(const v2f*)&L[ln16 * WL_STRIDE + ca];  // ds_load_b64
            acc2 = WMMA_F32(a, w2f[cc], acc2);
        }
        asm volatile("s_wait_dscnt 0" ::: "memory");   // WAR: reads before rewrite
        #pragma unroll
        for (int r = 0; r < 8; ++r)
            L[(r + (half << 3)) * WL_STRIDE + ln16] = swishf(acc2[r] + bias2);
        asm volatile("s_wait_dscnt 0" ::: "memory");

        if (half == 0) {                            // layer 3: 16 -> 1
            float a3 = b3v;
            #pragma unroll
            for (int c = 0; c < 16; ++c) a3 += L[ln16 * WL_STRIDE + c] * w3r[c];
            Wout[t0 + ln16] = a3;
        }
    }
}

// ---------------------------------------------------------------------------
// K6: in-place masked softmax over the contiguous axis of Wraw, then zero
// rows where mel_mask.  grid = B*K rows, block = 256 (2 elems / thread).
// ---------------------------------------------------------------------------
__global__ __launch_bounds__(256) void softmax_kernel(
    float* __restrict__ W, const int* __restrict__ src_mask, const float* __restrict__ mel)
{
    __shared__ float red[8];
    const int row = blockIdx.x, b = row >> 9, i = row & 511;
    float* Wr = W + (size_t)row * TT;
    const int tid = threadIdx.x, lane = tid & 31, wv = tid >> 5;

    const bool m0 = src_mask[b * KK + tid] != 0;
    const bool m1 = src_mask[b * KK + tid + 256] != 0;
    const float NEG = -__builtin_huge_valf();
    const float v0 = m0 ? NEG : Wr[tid];
    const float v1 = m1 ? NEG : Wr[tid + 256];

    float mx = wmax32(fmaxf(v0, v1));
    if (lane == 0) red[wv] = mx;
    __syncthreads();
    float m = red[0];
    #pragma unroll
    for (int j = 1; j < 8; ++j) m = fmaxf(m, red[j]);
    __syncthreads();

    const float e0 = m0 ? 0.f : expf(v0 - m);
    const float e1 = m1 ? 0.f : expf(v1 - m);
    float sm = wred32(e0 + e1);
    if (lane == 0) red[wv] = sm;
    __syncthreads();
    float tot = 0.f;
    #pragma unroll
    for (int j = 0; j < 8; ++j) tot += red[j];

    float scale = 1.f / tot;
    if (mel[b * TT + i] != 0.f) scale = 0.f;        // W = where(mel_mask, 0, W)
    Wr[tid] = e0 * scale;
    Wr[tid + 256] = e1 * scale;
}

// ---------------------------------------------------------------------------
// K7: right path fully fused: Craw computed on the fly (DC=4 MLPs), weighted
// by Wfinal, reduced cross-lane, projected through A -> rightA (B,T,256).
// One wave per output row.
// ---------------------------------------------------------------------------
__global__ __launch_bounds__(256) void right_kernel(
    const float* __restrict__ Wf, const float* __restrict__ V2,
    const float* __restrict__ svec, const float* __restrict__ dvec,
    const float* __restrict__ mel, const int* __restrict__ src_mask,
    const float* __restrict__ m1w, const float* __restrict__ m1b,
    const float* __restrict__ m2w, const float* __restrict__ m2b,
    const float* __restrict__ m3w, const float* __restrict__ m3b,
    const float* __restrict__ Amat, float* __restrict__ rightA)
{
    const int wv = threadIdx.x >> 5, lane = threadIdx.x & 31;
    const int row = blockIdx.x * 8 + wv;            // b*T + i  (raw row)
    const int b = row >> 9;

    float v2r[4], M1[16], M2[16], M3[16], B1[4], B2[4], B3[4];
    #pragma unroll
    for (int p = 0; p < 4; ++p) {
        v2r[p] = V2[(size_t)row * DCV + p];
        B1[p] = m1b[p]; B2[p] = m2b[p]; B3[p] = m3b[p];
    }
    #pragma unroll
    for (int p = 0; p < 16; ++p) { M1[p] = m1w[p]; M2[p] = m2w[p]; M3[p] = m3w[p]; }

    const float sk = svec[row], ek = sk + dvec[row];
    const int   srcm = src_mask[row];
    const float* Wr = Wf + (size_t)row * TT;
    const float* melb = mel + (size_t)b * TT;

    float acc[8] = {0.f, 0.f, 0.f, 0.f, 0.f, 0.f, 0.f, 0.f};
    #pragma unroll 1
    for (int j = lane; j < TT; j += 32) {
        const float wgt = Wr[j];
        const float pos = (float)(j + 1);
        const bool inv = srcm || (melb[j] != 0.f);
        const float se = inv ? 0.f : ((pos - sk) + (ek - pos));
        float g[4], h1[4], h2[4];
        #pragma unroll
        for (int c = 0; c < 4; ++c) g[c] = se + v2r[c];
        #pragma unroll
        for (int o = 0; o < 4; ++o) {
            float t1 = B1[o], t2 = B3[o];
            #pragma unroll
            for (int c = 0; c < 4; ++c) { t1 += g[c] * M1[c * 4 + o]; t2 += g[c] * M3[c * 4 + o]; }
            h1[o] = swishf(t1); h2[o] = swishf(t2);
        }
        #pragma unroll
        for (int o = 0; o < 4; ++o) {
            float t1 = B2[o], t2 = B3[o];
            #pragma unroll
            for (int c = 0; c < 4; ++c) { t1 += h1[c] * M2[c * 4 + o]; t2 += h2[c] * M3[c * 4 + o]; }
            acc[o]     += wgt * swishf(t1);         // C1
            acc[o + 4] += wgt * swishf(t2);         // C2 (mlp2_3 twice, as in source)
        }
    }
    #pragma unroll
    for (int p = 0; p < 8; ++p) acc[p] = wred32(acc[p]);

    for (int h = lane; h < 256; h += 32) {          // (1x8) @ A(8x256)
        float o = 0.f;
        #pragma unroll
        for (int p = 0; p < 8; ++p) o += acc[p] * Amat[p * 256 + h];
        rightA[(size_t)row * 256 + h] = o;
    }
}

// ---------------------------------------------------------------------------
// K8: left = Wfinal(512x512) @ V(512x256) via WMMA; epilogue adds rightA and
// applies the mel mask.  16 W rows staged in LDS with stride 514 (even ->
// aligned b64 A-loads; 514 % 64 != 0 -> conflict-free column fetch).
// ---------------------------------------------------------------------------
#define WS_STRIDE 514
__global__ __launch_bounds__(256) void left_kernel(
    const float* __restrict__ Wf, const float* __restrict__ V,
    const float* __restrict__ rightA, const float* __restrict__ mel,
    float* __restrict__ O)
{
    __shared__ float Ws[16 * WS_STRIDE];
    const int tid = threadIdx.x, wv = tid >> 5, lane = tid & 31;
    const int half = lane >> 4, ln16 = lane & 15;
    const int blk = blockIdx.x;                     // 64 blocks per batch
    const int b = blk >> 6, rr = blk & 63;
    const int i0 = (rr >> 1) << 4, hh = rr & 1;

    const float* Wb = Wf + ((size_t)b * TT + i0) * TT;
    for (int idx = tid; idx < 16 * 512; idx += 256) {
        const int row = idx >> 9, col = idx & 511;
        Ws[row * WS_STRIDE + col] = Wb[idx];
    }
    __syncthreads();

    const int n0 = hh * 128 + wv * 16;
    const float* Vb = V + (size_t)b * KK * 256;
    v8f acc = {};
    #pragma unroll 1
    for (int c0 = 0; c0 < 512; c0 += 4) {
        const int ca = c0 + (half << 1);
        if (c0 + 64 < 512)
            __builtin_prefetch(Vb + (size_t)(c0 + 64) * 256 + n0 + ln16, 0, 1);
        const v2f a = *(const v2f*)&Ws[ln16 * WS_STRIDE + ca];    // ds_load_b64
        v2f bb; bb.x = Vb[(size_t)ca * 256 + n0 + ln16];
                bb.y = Vb[(size_t)(ca + 1) * 256 + n0 + ln16];
        acc = WMMA_F32(a, bb, acc);
    }
    #pragma unroll
    for (int r = 0; r < 8; ++r) {
        const int i = i0 + r + (half << 3);
        const int h = n0 + ln16;
        const float melv = mel[b * TT + i];
        const float val = (melv != 0.f)
                        ? 0.f
                        : (acc[r] + rightA[((size_t)b * TT + i) * 256 + h]);
        O[((size_t)b * TT + i) * 256 + h] = val;
    }
}

// ---------------------------------------------------------------------------
// Host-side launch.
// Outputs (float32, concatenated): O (B,T,H) | d (B,K) | W (B,T,K) | mel (B,T)
// ---------------------------------------------------------------------------
extern "C" void kernel_launch(void* const* d_in, const int* in_sizes, int n_in,
                              void* d_out, int out_size, void* d_ws, size_t ws_size,
                              hipStream_t stream)
{
    (void)in_sizes; (void)n_in; (void)out_size; (void)ws_size;

    const float* x         = (const float*)d_in[0];
    const int*   src_mask  = (const int*)  d_in[1];
    const float* dp1_w     = (const float*)d_in[2];
    const float* dp1_b     = (const float*)d_in[3];
    const float* dp_ln1_g  = (const float*)d_in[4];
    const float* dp_ln1_b  = (const float*)d_in[5];
    const float* dp2_w     = (const float*)d_in[6];
    const float* dp2_b     = (const float*)d_in[7];
    const float* dp_ln2_g  = (const float*)d_in[8];
    const float* dp_ln2_b  = (const float*)d_in[9];
    const float* dp_lin_w  = (const float*)d_in[10];
    const float* dp_lin_b  = (const float*)d_in[11];
    const float* up_w_w    = (const float*)d_in[12];
    const float* up_w_b    = (const float*)d_in[13];
    const float* lnw_g     = (const float*)d_in[14];
    const float* lnw_b     = (const float*)d_in[15];
    const float* up_c_w    = (const float*)d_in[16];
    const float* up_c_b    = (const float*)d_in[17];
    const float* lnc_g     = (const float*)d_in[18];
    const float* lnc_b     = (const float*)d_in[19];
    const float* m11_w     = (const float*)d_in[20];
    const float* m11_b     = (const float*)d_in[21];
    const float* m12_w     = (const float*)d_in[22];
    const float* m12_b     = (const float*)d_in[23];
    const float* m13_w     = (const float*)d_in[24];
    const float* m13_b     = (const float*)d_in[25];
    const float* m21_w     = (const float*)d_in[26];
    const float* m21_b     = (const float*)d_in[27];
    const float* m22_w     = (const float*)d_in[28];
    const float* m22_b     = (const float*)d_in[29];
    const float* m23_w     = (const float*)d_in[30];
    const float* m23_b     = (const float*)d_in[31];
    const float* Amat      = (const float*)d_in[32];

    float* out = (float*)d_out;
    float* outO   = out;                    // 16*512*256 = 2097152
    float* outd   = out + 2097152;          // 16*512     =    8192
    float* outW   = out + 2105344;          // 16*512*512 = 4194304
    float* outmel = out + 6299648;          // 16*512     =    8192

    float* ws     = (float*)d_ws;
    float* wsH    = ws;                     // h       (B,K,256)
    float* wsV    = ws + 2097152;           // V       (B,K,256)
    float* wsV1   = ws + 4194304;           // V1      (B,K,16)
    float* wsV2   = ws + 4325376;           // V2      (B,K,4)
    float* wsS    = ws + 4358144;           // s       (B,K)
    float* wsRA   = ws + 4366336;           // rightA  (B,T,256)
    float* wsT1   = ws + 6463488;           // dp1_w^T (256,768)
    float* wsT2   = ws + 6660096;           // dp2_w^T (256,768)
    float* wsTw   = ws + 6856704;           // up_w^T  (16,768)

    // 0) one-shot weight transposes (b64-friendly B-fragment layout)
    transpose_w_kernel<<<(768 * 256 + 255) / 256, 256, 0, stream>>>(dp1_w, wsT1, 768, 256);
    transpose_w_kernel<<<(768 * 256 + 255) / 256, 256, 0, stream>>>(dp2_w, wsT2, 768, 256);
    transpose_w_kernel<<<(768 * 16 + 255) / 256, 256, 0, stream>>>(up_w_w, wsTw, 768, 16);

    // 1) h = LN(relu(conv1(x)))
    conv_relu_ln_kernel<false><<<BB * KK / 16, 512, 0, stream>>>(
        x, wsT1, dp1_b, dp_ln1_g, dp_ln1_b, wsH, nullptr, nullptr, nullptr, nullptr);
    // 2) V = LN(relu(conv2(h)));  d = masked softplus(V @ lin)
    conv_relu_ln_kernel<true><<<BB * KK / 16, 512, 0, stream>>>(
        wsH, wsT2, dp2_b, dp_ln2_g, dp_ln2_b, wsV, dp_lin_w, dp_lin_b, src_mask, outd);
    // 3) s = excl-cumsum(d); mel_mask from Tlen = rint(sum d)
    scan_kernel<<<BB, 512, 0, stream>>>(outd, wsS, outmel);
    // 4) V1, V2 branches
    upbranch_kernel<<<BB * KK / 16 / 8, 256, 0, stream>>>(
        wsV, wsTw, up_w_b, lnw_g, lnw_b, up_c_w, up_c_b, lnc_g, lnc_b, wsV1, wsV2);
    // 5) Wraw via swish-MLP grid (WMMA)
    wgrid_kernel<<<BB * KK / 8, 256, 0, stream>>>(
        wsV1, outd, wsS, outmel, src_mask, m11_w, m11_b, m12_w, m12_b, m13_w, m13_b, outW);
    // 6) masked softmax in place
    softmax_kernel<<<BB * KK, 256, 0, stream>>>(outW, src_mask, outmel);
    // 7) right path -> rightA
    right_kernel<<<BB * TT / 8, 256, 0, stream>>>(
        outW, wsV2, wsS, outd, outmel, src_mask,
        m21_w, m21_b, m22_w, m22_b, m23_w, m23_b, Amat, wsRA);
    // 8) left GEMM + epilogue -> O
    left_kernel<<<BB * 64, 256, 0, stream>>>(outW, wsV, wsRA, outmel, outO);
}